// MBOMAgent_3453153706194
// MI455X (gfx1250) — compile-verified
//
#include <hip/hip_runtime.h>
#include <math.h>

// Problem constants (from reference)
#define BS    4096
#define NAG   8
#define HID   256
#define NACT  16
#define MESS  64
#define STEPS 5
#define INP   128
#define TRJ   (STEPS * NACT)   // 80
#define LAMQ  0.5f
#define R     (BS * NAG)       // 32768 rows

typedef float v2f __attribute__((ext_vector_type(2)));
typedef float v8f __attribute__((ext_vector_type(8)));

// CDNA5 fp32 WMMA: D(16x16,f32) = A(16x4,f32) x B(4x16,f32) + C
__device__ __forceinline__ v8f wmma4(v2f a, v2f b, v8f c) {
  return __builtin_amdgcn_wmma_f32_16x16x4_f32(
      /*neg_a=*/false, a, /*neg_b=*/false, b,
      /*c_mod=*/(short)0, c, /*reuse_a=*/false, /*reuse_b=*/false);
}

// Accumulate C(16x16) += A[16 x kLen] * W[16 x kLen]^T
// aBase -> A[m0][0] (leading dim lda), wBase -> W[n0][0] (leading dim ldw).
// kLen must be a multiple of 4; lda/ldw must be even (8B-aligned float2 loads).
// A-frag layout: lanes 0-15 hold (M=lane, K=k+0..1), lanes 16-31 hold (M, K=k+2..3).
// B-frag layout: lanes 0-15 hold (N=lane, K=k+0..1) of W row n0+lane, etc.
__device__ __forceinline__ void gemm_acc(const float* __restrict__ aBase, int lda,
                                         const float* __restrict__ wBase, int ldw,
                                         int kLen, int lane, v8f& c) {
  const int half = lane >> 4;
  const int r16  = lane & 15;
  const float* ap = aBase + r16 * lda + 2 * half;
  const float* wp = wBase + r16 * ldw + 2 * half;
  for (int k = 0; k < kLen; k += 4) {
    v2f a = *(const v2f*)(ap + k);
    v2f b = *(const v2f*)(wp + k);
    c = wmma4(a, b, c);
  }
}

// ---------------- Kernel 1: X = relu(in @ fc1_w^T + fc1_b) ----------------
__global__ void k_fc1(const float* __restrict__ in, const float* __restrict__ w,
                      const float* __restrict__ b, float* __restrict__ X) {
  const int wave = (blockIdx.x * blockDim.x + threadIdx.x) >> 5;
  const int lane = threadIdx.x & 31;
  const int nT = HID / 16;
  const int m0 = (wave / nT) * 16;
  const int n0 = (wave % nT) * 16;
  v8f c = {};
  gemm_acc(in + (size_t)m0 * INP, INP, w + (size_t)n0 * INP, INP, INP, lane, c);
  const int half = lane >> 4, col = lane & 15;
  const float bv = b[n0 + col];
#pragma unroll
  for (int r = 0; r < 8; ++r) {
    float v = c[r] + bv;
    v = v > 0.f ? v : 0.f;
    X[(size_t)(m0 + r + 8 * half) * HID + n0 + col] = v;
  }
}

// ---------------- Kernel 2: GRU cell, fused 6 gate tiles ----------------
__global__ void k_gru(const float* __restrict__ X, const float* __restrict__ hOld,
                      const float* __restrict__ wih, const float* __restrict__ whh,
                      const float* __restrict__ bih, const float* __restrict__ bhh,
                      float* __restrict__ hOut) {
  const int wave = (blockIdx.x * blockDim.x + threadIdx.x) >> 5;
  const int lane = threadIdx.x & 31;
  const int nT = HID / 16;
  const int m0 = (wave / nT) * 16;
  const int n0 = (wave % nT) * 16;
  const int half = lane >> 4, r16 = lane & 15;

  v8f cir = {}, ciz = {}, cin = {}, chr_ = {}, chz = {}, chn = {};
  const float* ax = X    + (size_t)(m0 + r16) * HID + 2 * half;
  const float* ah = hOld + (size_t)(m0 + r16) * HID + 2 * half;
  const float* wr = wih + (size_t)(0 * HID + n0 + r16) * HID + 2 * half;
  const float* wz = wih + (size_t)(1 * HID + n0 + r16) * HID + 2 * half;
  const float* wn = wih + (size_t)(2 * HID + n0 + r16) * HID + 2 * half;
  const float* ur = whh + (size_t)(0 * HID + n0 + r16) * HID + 2 * half;
  const float* uz = whh + (size_t)(1 * HID + n0 + r16) * HID + 2 * half;
  const float* un = whh + (size_t)(2 * HID + n0 + r16) * HID + 2 * half;

  for (int k = 0; k < HID; k += 4) {
    v2f a0 = *(const v2f*)(ax + k);
    v2f a1 = *(const v2f*)(ah + k);
    cir  = wmma4(a0, *(const v2f*)(wr + k), cir);
    ciz  = wmma4(a0, *(const v2f*)(wz + k), ciz);
    cin  = wmma4(a0, *(const v2f*)(wn + k), cin);
    chr_ = wmma4(a1, *(const v2f*)(ur + k), chr_);
    chz  = wmma4(a1, *(const v2f*)(uz + k), chz);
    chn  = wmma4(a1, *(const v2f*)(un + k), chn);
  }

  const int col = lane & 15;
  const float bir = bih[n0 + col], biz = bih[HID + n0 + col], bin = bih[2 * HID + n0 + col];
  const float bhr = bhh[n0 + col], bhz = bhh[HID + n0 + col], bhn = bhh[2 * HID + n0 + col];
#pragma unroll
  for (int r = 0; r < 8; ++r) {
    const int row = m0 + r + 8 * half;
    const float ir = cir[r] + bir, iz = ciz[r] + biz, inn = cin[r] + bin;
    const float hr = chr_[r] + bhr, hz = chz[r] + bhz, hn = chn[r] + bhn;
    const float rg = 1.f / (1.f + __expf(-(ir + hr)));
    const float zg = 1.f / (1.f + __expf(-(iz + hz)));
    const float ng = tanhf(inn + rg * hn);
    const float ho = hOld[(size_t)row * HID + n0 + col];
    hOut[(size_t)row * HID + n0 + col] = (1.f - zg) * ng + zg * ho;
  }
}

// ---------------- Kernel 3a: q_inf = h @ dec^T + b; traj columns; argmax actions ----
__global__ void k_qinf(const float* __restrict__ h, const float* __restrict__ decw,
                       const float* __restrict__ decb, float* __restrict__ traj,
                       float* __restrict__ ac, int s) {
  __shared__ float tile[4][16][16];
  const int wslot = threadIdx.x >> 5;
  const int wave = (blockIdx.x * blockDim.x + threadIdx.x) >> 5;
  const int lane = threadIdx.x & 31;
  const int m0 = wave * 16;
  v8f c = {};
  gemm_acc(h + (size_t)m0 * HID, HID, decw, HID, HID, lane, c);
  const int half = lane >> 4, col = lane & 15;
  const float bv = decb[col];
#pragma unroll
  for (int r = 0; r < 8; ++r) {
    const float v = c[r] + bv;
    traj[(size_t)(m0 + r + 8 * half) * TRJ + s * NACT + col] = v;
    tile[wslot][r + 8 * half][col] = v;
  }
  __syncthreads();
  if (lane < 16) {
    float best = tile[wslot][lane][0];
    int bi = 0;
#pragma unroll
    for (int j = 1; j < NACT; ++j) {
      const float v = tile[wslot][lane][j];
      if (v > best) { best = v; bi = j; }   // keep first max (jnp.argmax semantics)
    }
    ac[m0 + lane] = (float)bi;
  }
}

// ---------------- Kernel 3b: h_next = [h, broadcast(actions)] @ trans^T + b --------
__global__ void k_trans(const float* __restrict__ h, const float* __restrict__ ac,
                        const float* __restrict__ tw, const float* __restrict__ tb,
                        float* __restrict__ hNext) {
  const int wave = (blockIdx.x * blockDim.x + threadIdx.x) >> 5;
  const int lane = threadIdx.x & 31;
  const int nT = HID / 16;
  const int m0 = (wave / nT) * 16;
  const int n0 = (wave % nT) * 16;
  const int half = lane >> 4, r16 = lane & 15;
  const int ldw = HID + NAG;  // 264

  v8f c = {};
  // hc part: K = 0..255
  gemm_acc(h + (size_t)m0 * HID, HID, tw + (size_t)n0 * ldw, ldw, HID, lane, c);
  // action part: K = 256..263, A[row][256+j] = ac[(row & ~7) + j]
  {
    const float* ap = ac + (((m0 + r16) & ~7)) + 2 * half;
    const float* wp = tw + (size_t)(n0 + r16) * ldw + HID + 2 * half;
#pragma unroll
    for (int k = 0; k < NAG; k += 4)
      c = wmma4(*(const v2f*)(ap + k), *(const v2f*)(wp + k), c);
  }
  const int col = lane & 15;
  const float bv = tb[n0 + col];
#pragma unroll
  for (int r = 0; r < 8; ++r)
    hNext[(size_t)(m0 + r + 8 * half) * HID + n0 + col] = c[r] + bv;
}

// ---------------- Kernel 4: tiny 8-agent attention per batch ----------------
__global__ void k_attn(const float* __restrict__ traj,
                       const float* __restrict__ qw, const float* __restrict__ qb,
                       const float* __restrict__ kw, const float* __restrict__ kb,
                       const float* __restrict__ vw, const float* __restrict__ vb,
                       float* __restrict__ msg) {
  __shared__ float qkv[2][16][192];  // per-wave: cols 0-63 Q, 64-127 K, 128-191 V
  const int wslot = threadIdx.x >> 5;
  const int wave = (blockIdx.x * blockDim.x + threadIdx.x) >> 5;
  const int lane = threadIdx.x & 31;
  const int m0 = wave * 16;  // 16 rows = 2 batches of 8 agents
  const int half = lane >> 4, col = lane & 15;

  for (int nt = 0; nt < 12; ++nt) {
    const float* W; const float* B; int n0;
    if (nt < 4)      { W = qw; B = qb; n0 = nt * 16; }
    else if (nt < 8) { W = kw; B = kb; n0 = (nt - 4) * 16; }
    else             { W = vw; B = vb; n0 = (nt - 8) * 16; }
    v8f c = {};
    gemm_acc(traj + (size_t)m0 * TRJ, TRJ, W + (size_t)n0 * TRJ, TRJ, TRJ, lane, c);
    const float bv = B[n0 + col];
#pragma unroll
    for (int r = 0; r < 8; ++r)
      qkv[wslot][r + 8 * half][nt * 16 + col] = c[r] + bv;
  }
  __syncthreads();

  if (lane < 16) {
    const int bl = lane >> 3;   // local batch (0/1)
    const int n  = lane & 7;    // agent
    const float (*T)[192] = qkv[wslot];
    float sc[8];
    float mx = -1e30f;
#pragma unroll
    for (int m = 0; m < 8; ++m) {
      float s = 0.f;
      for (int d = 0; d < MESS; ++d) s += T[bl * 8 + n][d] * T[bl * 8 + m][64 + d];
      s *= 0.125f;                 // 1/sqrt(MESS)
      sc[m] = s;
      mx = s > mx ? s : mx;
    }
    float sum = 0.f;
#pragma unroll
    for (int m = 0; m < 8; ++m) { sc[m] = __expf(sc[m] - mx); sum += sc[m]; }
    const float inv = 1.f / sum;
    for (int d = 0; d < MESS; ++d) {
      float acc = 0.f;
#pragma unroll
      for (int m = 0; m < 8; ++m) acc += sc[m] * T[bl * 8 + m][128 + d];
      msg[(size_t)(m0 + lane) * MESS + d] = acc * inv;
    }
  }
}

// ---------------- Kernel 5: q = (h@dec^T+b) + 0.5*([h,msg]@qij^T+b) ----------------
__global__ void k_qout(const float* __restrict__ h, const float* __restrict__ msg,
                       const float* __restrict__ decw, const float* __restrict__ decb,
                       const float* __restrict__ qijw, const float* __restrict__ qijb,
                       float* __restrict__ q) {
  const int wave = (blockIdx.x * blockDim.x + threadIdx.x) >> 5;
  const int lane = threadIdx.x & 31;
  const int m0 = wave * 16;
  const int ldq = HID + MESS;  // 320
  v8f c1 = {}, c2 = {};
  gemm_acc(h + (size_t)m0 * HID, HID, decw, HID, HID, lane, c1);
  gemm_acc(h + (size_t)m0 * HID, HID, qijw, ldq, HID, lane, c2);
  gemm_acc(msg + (size_t)m0 * MESS, MESS, qijw + HID, ldq, MESS, lane, c2);
  const int half = lane >> 4, col = lane & 15;
  const float b1 = decb[col], b2 = qijb[col];
#pragma unroll
  for (int r = 0; r < 8; ++r)
    q[(size_t)(m0 + r + 8 * half) * NACT + col] = (c1[r] + b1) + LAMQ * (c2[r] + b2);
}

extern "C" void kernel_launch(void* const* d_in, const int* in_sizes, int n_in,
                              void* d_out, int out_size, void* d_ws, size_t ws_size,
                              hipStream_t stream) {
  const float* inputs = (const float*)d_in[0];
  const float* hold   = (const float*)d_in[1];
  const float* fc1w   = (const float*)d_in[2];
  const float* fc1b   = (const float*)d_in[3];
  const float* wih    = (const float*)d_in[4];
  const float* whh    = (const float*)d_in[5];
  const float* bih    = (const float*)d_in[6];
  const float* bhh    = (const float*)d_in[7];
  const float* decw   = (const float*)d_in[8];
  const float* decb   = (const float*)d_in[9];
  const float* tw     = (const float*)d_in[10];
  const float* tb     = (const float*)d_in[11];
  const float* qijw   = (const float*)d_in[12];
  const float* qijb   = (const float*)d_in[13];
  const float* qw     = (const float*)d_in[14];
  const float* qb     = (const float*)d_in[15];
  const float* kw     = (const float*)d_in[16];
  const float* kb     = (const float*)d_in[17];
  const float* vw     = (const float*)d_in[18];
  const float* vb     = (const float*)d_in[19];

  float* qOut = (float*)d_out;                       // (BS,NAG,NACT)
  float* hOut = (float*)d_out + (size_t)R * NACT;    // (BS,NAG,HID)

  // Workspace layout (floats):
  //   [0 .. R*HID)          : X (fc1 output), reused as h ping buffer 0
  //   [R*HID .. 2*R*HID)    : h ping buffer 1
  //   then traj (R*TRJ), ac (R), msg (R*MESS)
  float* ws   = (float*)d_ws;
  float* X    = ws;
  float* hp0  = ws;                       // X is dead after k_gru
  float* hp1  = ws + (size_t)R * HID;
  float* traj = ws + (size_t)2 * R * HID;
  float* ac   = traj + (size_t)R * TRJ;
  float* msg  = ac + R;

  const int gemmTiles = (R / 16) * (HID / 16);  // 32768 waves

  k_fc1<<<gemmTiles / 4, 128, 0, stream>>>(inputs, fc1w, fc1b, X);
  k_gru<<<gemmTiles / 4, 128, 0, stream>>>(X, hold, wih, whh, bih, bhh, hOut);

  // Imagination rollout: h0 = hOut (stop_gradient is identity in fwd).
  float* hs[STEPS] = { hOut, hp0, hp1, hp0, hp1 };
  for (int s = 0; s < STEPS; ++s) {
    k_qinf<<<(R / 16) / 4, 128, 0, stream>>>(hs[s], decw, decb, traj, ac, s);
    if (s < STEPS - 1)
      k_trans<<<gemmTiles / 4, 128, 0, stream>>>(hs[s], ac, tw, tb, hs[s + 1]);
  }

  k_attn<<<(R / 16) / 2, 64, 0, stream>>>(traj, qw, qb, kw, kb, vw, vb, msg);
  k_qout<<<(R / 16) / 4, 128, 0, stream>>>(hOut, msg, decw, decb, qijw, qijb, qOut);
}